// SimCLRLossV_11381663334552
// MI455X (gfx1250) — compile-verified
//
#include <hip/hip_runtime.h>
#include <hip/hip_bf16.h>

// ---------------------------------------------------------------------------
// SimCLR variance-weighted loss for MI455X (gfx1250, wave32, WMMA).
//
// dist[i,j] = sum_d (x_i - x_j)^2 / sigma_j  fused into ONE fp32 GEMM:
//   A'[i,:] = [ x_i^2 | x_i ]          (8192 x 1024)
//   B'[j,:] = [ 1/s_j | -2*x_j/s_j ]   (8192 x 1024)
//   dist    = A' B'^T + t3[j]
// logit_mat = 268 MB fp32 -> store-bound floor ~11.5 us @ 23.3 TB/s.
// GEMM = 137 GFLOP fp32 via V_WMMA_F32_16X16X4_F32; A'+B' (64 MB) stay hot
// in the 192 MB L2 because the output stream is written with NT temporal
// hints (__builtin_nontemporal_store).  K loop is software-pipelined with
// double-buffered fragments; all 6 load streams share 2 base pointers with
// constant byte offsets (r/t * 65536 fits the signed 24-bit IOFFSET), which
// minimizes address VGPRs so the scheduler keeps partial s_wait_loadcnt.
// Wave tile = 32x64 (8 WMMA per 6 loads per k-step), workgroup tile 64x256.
// ---------------------------------------------------------------------------

typedef float v2f __attribute__((ext_vector_type(2)));
typedef float v8f __attribute__((ext_vector_type(8)));

#define B2  8192
#define D   512
#define KP  1024            // packed K = 2*D
#define NM1 (B2 - 1)
#define RSTRIDE (16 * KP * 4)   // byte stride between 16-row groups = 65536

static __device__ __forceinline__ v8f wmma_f32(v2f a, v2f b, v8f c) {
    return __builtin_amdgcn_wmma_f32_16x16x4_f32(false, a, false, b, (short)0, c,
                                                 false, false);
}

// --------------------------------------------------------------------------
// Kernel 1: build A', B', per-row t3 / ls, zero rowsum. One block per row.
// --------------------------------------------------------------------------
__global__ __launch_bounds__(256)
void simclrv_prep(const float* __restrict__ feat, const float* __restrict__ sigma,
                  float* __restrict__ Ap, float* __restrict__ Bp,
                  float* __restrict__ t3, float* __restrict__ ls,
                  float* __restrict__ rowsum) {
    const int i   = blockIdx.x;
    const int tid = threadIdx.x;
    const int d   = tid * 2;                        // 256 threads * 2 = 512
    __shared__ float r3[256];
    __shared__ float rl[256];

    const v2f x = *(const v2f*)(feat  + (size_t)i * D + d);
    const v2f s = *(const v2f*)(sigma + (size_t)i * D + d);
    v2f invs, xx, nxs;
    invs.x = 1.0f / s.x;            invs.y = 1.0f / s.y;
    xx.x   = x.x * x.x;             xx.y   = x.y * x.y;
    nxs.x  = -2.0f * x.x * invs.x;  nxs.y  = -2.0f * x.y * invs.y;

    *(v2f*)(Ap + (size_t)i * KP + d)     = xx;
    *(v2f*)(Ap + (size_t)i * KP + D + d) = x;
    *(v2f*)(Bp + (size_t)i * KP + d)     = invs;
    *(v2f*)(Bp + (size_t)i * KP + D + d) = nxs;

    r3[tid] = xx.x * invs.x + xx.y * invs.y;
    rl[tid] = logf(s.x) + logf(s.y);
    __syncthreads();
    for (int off = 128; off > 0; off >>= 1) {
        if (tid < off) { r3[tid] += r3[tid + off]; rl[tid] += rl[tid + off]; }
        __syncthreads();
    }
    if (tid == 0) { t3[i] = r3[0]; ls[i] = rl[0]; rowsum[i] = 0.0f; }
}

// --------------------------------------------------------------------------
// Kernel 2: gt[i] = dist[i, pos_idx[i]] = A'[i].B'[p] + t3[p]; lsp[i] = ls[p]
// One wave32 per row (8 waves / block).
// --------------------------------------------------------------------------
__global__ __launch_bounds__(256)
void simclrv_gt(const int* __restrict__ label,
                const float* __restrict__ Ap, const float* __restrict__ Bp,
                const float* __restrict__ t3, const float* __restrict__ ls,
                float* __restrict__ gt, float* __restrict__ lsp) {
    const int wave = threadIdx.x >> 5;
    const int lane = threadIdx.x & 31;
    const int i    = blockIdx.x * 8 + wave;
    if (i >= B2) return;
    const int lbl = label[i];
    const int p   = lbl + (lbl >= i ? 1 : 0);       // absolute positive index

    const float* a = Ap + (size_t)i * KP;
    const float* b = Bp + (size_t)p * KP;
    float acc = 0.0f;
    #pragma unroll 4
    for (int k = lane * 2; k < KP; k += 64) {       // b64 loads per lane
        v2f av = *(const v2f*)(a + k);
        v2f bv = *(const v2f*)(b + k);
        acc += av.x * bv.x + av.y * bv.y;
    }
    for (int m = 16; m > 0; m >>= 1) acc += __shfl_xor(acc, m, 32);
    if (lane == 0) { gt[i] = acc + t3[p]; lsp[i] = ls[p]; }
}

// --------------------------------------------------------------------------
// Kernel 3: fused GEMM + epilogue.
// Block = 256 threads = 8 waves, 2 row groups x 4 col groups.
// Each wave: 32x64 strip = 8 v8f accumulators; K pipelined (double-buffered).
// fp32 WMMA fragment layout (ISA 7.12.2):
//   A 16x4 : lane -> M = lane%16,  VGPR{0,1} = K {0,1} (+2 for lanes>=16)
//   B 4x16 : lane -> N = lane%16,  VGPR{0,1} = K {0,1} (+2 for lanes>=16)
//   C 16x16: lane -> N = lane%16,  VGPR v    = M = v + 8*(lane>=16)
// --------------------------------------------------------------------------
__global__ __launch_bounds__(256)
void simclrv_dist(const float* __restrict__ Ap, const float* __restrict__ Bp,
                  const float* __restrict__ t3, const float* __restrict__ ls,
                  const float* __restrict__ lsp, const float* __restrict__ gt,
                  float* __restrict__ rowsum, float* __restrict__ out) {
    const int lane = threadIdx.x & 31;
    const int wave = threadIdx.x >> 5;
    const int wr   = wave >> 2;               // 0..1
    const int wc   = wave & 3;                // 0..3
    const int rowBase = blockIdx.y * 64  + wr * 32;
    const int colBase = blockIdx.x * 256 + wc * 64;

    const int half = (lane < 16) ? 0 : 1;     // K sub-pair select
    const int l16  = lane & 15;

    // Two base pointers; row/col sub-groups addressed via constant byte
    // offsets (r,t)*RSTRIDE which fit the signed 24-bit instruction offset.
    const char* ab = (const char*)(Ap + (size_t)(rowBase + l16) * KP) + half * 8;
    const char* bb = (const char*)(Bp + (size_t)(colBase + l16) * KP) + half * 8;
#define AFRAG(r, kb) (*(const v2f*)(ab + (r) * RSTRIDE + (kb)))
#define BFRAG(t, kb) (*(const v2f*)(bb + (t) * RSTRIDE + (kb)))

    __builtin_prefetch(ab, 0, 1);
    __builtin_prefetch(bb, 0, 1);
    __builtin_prefetch(bb + 2 * RSTRIDE, 0, 1);

    v8f c[2][4];
    #pragma unroll
    for (int r = 0; r < 2; ++r)
        #pragma unroll
        for (int t = 0; t < 4; ++t) c[r][t] = (v8f){};

    // ---- software-pipelined K loop (byte offsets, 16 B per k-step of 4) ----
    v2f ac[2], bc[4], an[2], bn[4];
    #pragma unroll
    for (int r = 0; r < 2; ++r) ac[r] = AFRAG(r, 0);
    #pragma unroll
    for (int t = 0; t < 4; ++t) bc[t] = BFRAG(t, 0);

    #pragma unroll 4
    for (int kb = 16; kb < KP * 4; kb += 16) {
        #pragma unroll
        for (int r = 0; r < 2; ++r) an[r] = AFRAG(r, kb);
        #pragma unroll
        for (int t = 0; t < 4; ++t) bn[t] = BFRAG(t, kb);

        #pragma unroll
        for (int r = 0; r < 2; ++r)
            #pragma unroll
            for (int t = 0; t < 4; ++t)
                c[r][t] = wmma_f32(ac[r], bc[t], c[r][t]);

        #pragma unroll
        for (int r = 0; r < 2; ++r) ac[r] = an[r];
        #pragma unroll
        for (int t = 0; t < 4; ++t) bc[t] = bn[t];
    }
    #pragma unroll
    for (int r = 0; r < 2; ++r)
        #pragma unroll
        for (int t = 0; t < 4; ++t)
            c[r][t] = wmma_f32(ac[r], bc[t], c[r][t]);
#undef AFRAG
#undef BFRAG

    // ---- epilogue: dist -> logit_mat (skip-self gather) + row sums ----
    float t3v[4], lpv[4]; int jv[4];
    #pragma unroll
    for (int t = 0; t < 4; ++t) {
        const int j = colBase + t * 16 + l16;
        jv[t] = j; t3v[t] = t3[j]; lpv[t] = lsp[j];
    }

    #pragma unroll
    for (int r = 0; r < 2; ++r) {
        #pragma unroll
        for (int v = 0; v < 8; ++v) {
            const int   i   = rowBase + r * 16 + v + half * 8;
            const float lsi = ls[i];
            const float gti = gt[i];
            float rs = 0.0f;
            #pragma unroll
            for (int t = 0; t < 4; ++t) {
                const int j = jv[t];
                if (j == i) continue;                   // skip self column
                const float dist = c[r][t][v] + t3v[t];
                const float val  = expf(0.5f * (lsi - lpv[t]) + 0.1f * (gti - dist));
                rs += val;
                const int jj = j - (j > i ? 1 : 0);     // gathered column
                // NT store: stream 268 MB out without evicting L2-resident A'/B'
                __builtin_nontemporal_store(val, &out[(size_t)i * NM1 + jj]);
            }
            // reduce rs across the 16 lanes of this half (same row i)
            for (int m = 1; m < 16; m <<= 1) rs += __shfl_xor(rs, m, 32);
            if (l16 == 0) atomicAdd(&rowsum[i], rs);
        }
    }
}

// --------------------------------------------------------------------------
// Kernel 4: loss = mean(log(rowsum)). Single block.
// --------------------------------------------------------------------------
__global__ __launch_bounds__(1024)
void simclrv_loss(const float* __restrict__ rowsum, float* __restrict__ loss) {
    __shared__ float red[1024];
    const int tid = threadIdx.x;
    float acc = 0.0f;
    for (int i = tid; i < B2; i += 1024) acc += logf(rowsum[i]);
    red[tid] = acc;
    __syncthreads();
    for (int off = 512; off > 0; off >>= 1) {
        if (tid < off) red[tid] += red[tid + off];
        __syncthreads();
    }
    if (tid == 0) *loss = red[0] / (float)B2;
}

// --------------------------------------------------------------------------
extern "C" void kernel_launch(void* const* d_in, const int* in_sizes, int n_in,
                              void* d_out, int out_size, void* d_ws, size_t ws_size,
                              hipStream_t stream) {
    const float* feat  = (const float*)d_in[0];
    const float* sigma = (const float*)d_in[1];
    const int*   label = (const int*)  d_in[2];

    float* out   = (float*)d_out;     // out[0] = loss, out[1..] = logit_mat
    float* loss  = out;
    float* logit = out + 1;

    // workspace (floats): Ap | Bp | t3 | ls | lsp | gt | rowsum  (~64.2 MB)
    float* ws     = (float*)d_ws;
    float* Ap     = ws;
    float* Bp     = Ap + (size_t)B2 * KP;
    float* t3     = Bp + (size_t)B2 * KP;
    float* ls     = t3 + B2;
    float* lsp    = ls + B2;
    float* gt     = lsp + B2;
    float* rowsum = gt + B2;

    simclrv_prep<<<B2, 256, 0, stream>>>(feat, sigma, Ap, Bp, t3, ls, rowsum);
    simclrv_gt  <<<B2 / 8, 256, 0, stream>>>(label, Ap, Bp, t3, ls, gt, lsp);
    dim3 grid(B2 / 256, B2 / 64);     // (32, 128) workgroups, 256 threads each
    simclrv_dist<<<grid, 256, 0, stream>>>(Ap, Bp, t3, ls, lsp, gt, rowsum, logit);
    simclrv_loss<<<1, 1024, 0, stream>>>(rowsum, loss);
}